// GAT_40295383171397
// MI455X (gfx1250) — compile-verified
//
#include <hip/hip_runtime.h>
#include <math.h>

// ---------------------------------------------------------------------------
// GAT (2-layer, PyG GATConv semantics) for MI455X / gfx1250.
//  - GEMMs via V_WMMA_F32_16X16X4_F32 (full fp32: workload is scatter-bound,
//    so no precision tradeoff buys wall clock).
//  - Edge softmax via monotone-uint atomicMax + f32 atomicAdd segment ops.
// Workspace use: ~168 MB (< typical ws). All async, graph-capture safe.
// ---------------------------------------------------------------------------

typedef __attribute__((ext_vector_type(2))) float v2f;
typedef __attribute__((ext_vector_type(8))) float v8f;

__device__ __forceinline__ unsigned ford(float v) {
  unsigned u = __float_as_uint(v);
  return (u & 0x80000000u) ? ~u : (u | 0x80000000u);
}
__device__ __forceinline__ float unford(unsigned u) {
  u = (u & 0x80000000u) ? (u & 0x7fffffffu) : ~u;
  return __uint_as_float(u);
}
__device__ __forceinline__ float elu1(float x)  { return x > 0.f ? x : expm1f(x); }
__device__ __forceinline__ float lrelu(float x) { return x > 0.f ? x : 0.2f * x; }

__device__ __forceinline__ void edge_sd(const int* __restrict__ ei, int E, int e,
                                        int& s, int& d) {
  if (e < E) { s = ei[e]; d = ei[E + e]; }
  else       { s = e - E; d = s; }          // appended self-loops
}

// --------------------------- WMMA fp32 GEMM --------------------------------
// C[M,N] = A[M,K] @ B[K,N], row-major, K%4==0, N%16==0. One 16x16 tile/wave.
__global__ void __launch_bounds__(256) wmma_gemm_f32(
    const float* __restrict__ A, const float* __restrict__ B,
    float* __restrict__ C, int M, int N, int K, int tiles_n, int total_tiles) {
  const int wave = threadIdx.x >> 5;
  const int lane = threadIdx.x & 31;
  const int tile = blockIdx.x * 8 + wave;
  if (tile >= total_tiles) return;                 // wave-uniform: EXEC stays full
  const int mt = tile / tiles_n;
  const int nt = tile - mt * tiles_n;
  const int lm = lane & 15;
  const int kh = lane >> 4;                        // 0/1: K-pair half
  int mrow = (mt << 4) + lm;
  if (mrow >= M) mrow = M - 1;                     // clamp (keeps EXEC all-1s)
  const int ncol = (nt << 4) + lm;

  const float* Ap = A + (size_t)mrow * K + (kh << 1);
  const float* Bp = B + (size_t)(kh << 1) * N + ncol;

  v8f acc = {0.f, 0.f, 0.f, 0.f, 0.f, 0.f, 0.f, 0.f};
  for (int k0 = 0; k0 < K; k0 += 4) {
    v2f a = *(const v2f*)Ap;                       // A[mrow][k0+2kh +0/+1] (8B aligned)
    v2f b;
    b.x = Bp[0];                                   // B[k0+2kh+0][ncol]
    b.y = Bp[N];                                   // B[k0+2kh+1][ncol]
    acc = __builtin_amdgcn_wmma_f32_16x16x4_f32(
        /*neg_a=*/false, a, /*neg_b=*/false, b,
        /*c_mod=*/(short)0, acc, /*reuse_a=*/false, /*reuse_b=*/false);
    Ap += 4;
    Bp += (size_t)4 * N;
  }
  const int mbase = (mt << 4) + (kh << 3);         // vgpr v -> row mbase+v
#pragma unroll
  for (int v = 0; v < 8; ++v) {
    const int r = mbase + v;
    if (r < M) C[(size_t)r * N + ncol] = acc[v];
  }
}

// --------------------------- pipeline kernels ------------------------------
__global__ void fill_u32(unsigned* __restrict__ p, unsigned val, int n) {
  int t = blockIdx.x * blockDim.x + threadIdx.x;
  if (t < n) p[t] = val;
}

// h[n,0:128] = [ent_emb[x[n]], rel_emb[et[n]]]
__global__ void build_h(const int* __restrict__ x, const int* __restrict__ et,
                        const float* __restrict__ ent_emb,
                        const float* __restrict__ rel_emb,
                        float* __restrict__ h) {
  const int n = blockIdx.x, c = threadIdx.x;       // blockDim = 128
  float v = (c < 64) ? ent_emb[(size_t)x[n] * 64 + c]
                     : rel_emb[(size_t)et[n] * 64 + (c - 64)];
  h[(size_t)n * 128 + c] = v;
}

// per-(node,head) attention logits, layer 1 (H=8, C=64)
__global__ void logits1(const float* __restrict__ hl, const float* __restrict__ as_,
                        const float* __restrict__ ad_, float* __restrict__ als,
                        float* __restrict__ ald, int N) {
  int t = blockIdx.x * blockDim.x + threadIdx.x;
  if (t >= N * 8) return;
  const int n = t >> 3, hd = t & 7;
  const float* row = hl + (size_t)n * 512 + hd * 64;
  const float* pa = as_ + hd * 64;
  const float* pb = ad_ + hd * 64;
  float ss = 0.f, sd = 0.f;
#pragma unroll 8
  for (int c = 0; c < 64; ++c) { float v = row[c]; ss += v * pa[c]; sd += v * pb[c]; }
  als[t] = ss; ald[t] = sd;
}

__global__ void edge1_logits(const int* __restrict__ ei, int E, int E2,
                             const float* __restrict__ als, const float* __restrict__ ald,
                             float* __restrict__ eb, unsigned* __restrict__ m1) {
  int e = blockIdx.x * blockDim.x + threadIdx.x;
  if (e >= E2) return;
  int s, d; edge_sd(ei, E, e, s, d);
#pragma unroll
  for (int hd = 0; hd < 8; ++hd) {
    float v = lrelu(als[s * 8 + hd] + ald[d * 8 + hd]);
    eb[(size_t)e * 8 + hd] = v;
    atomicMax(&m1[d * 8 + hd], ford(v));
  }
}

__global__ void edge1_expsum(const int* __restrict__ ei, int E, int E2,
                             float* __restrict__ eb, const unsigned* __restrict__ m1,
                             float* __restrict__ s1) {
  int e = blockIdx.x * blockDim.x + threadIdx.x;
  if (e >= E2) return;
  int s, d; edge_sd(ei, E, e, s, d);
#pragma unroll
  for (int hd = 0; hd < 8; ++hd) {
    float ex = __expf(eb[(size_t)e * 8 + hd] - unford(m1[d * 8 + hd]));
    eb[(size_t)e * 8 + hd] = ex;
    atomicAdd(&s1[d * 8 + hd], ex);
  }
}

// one 256-thread block per edge: 512 channels, 2/thread
__global__ void __launch_bounds__(256) edge1_agg(
    const int* __restrict__ ei, int E, const float* __restrict__ eb,
    const float* __restrict__ s1, const float* __restrict__ hl,
    float* __restrict__ agg1) {
  const int e = blockIdx.x;
  int s, d; edge_sd(ei, E, e, s, d);
  const int idx = threadIdx.x * 2;
  const int hd = idx >> 6;
  const float alpha = eb[(size_t)e * 8 + hd] / (s1[d * 8 + hd] + 1e-16f);
  v2f hv = *(const v2f*)(hl + (size_t)s * 512 + idx);
  atomicAdd(&agg1[(size_t)d * 512 + idx + 0], hv.x * alpha);
  atomicAdd(&agg1[(size_t)d * 512 + idx + 1], hv.y * alpha);
}

// h2[n,0:576] = [elu(agg1[n]+b1), ent_emb[x[n]]]
__global__ void build_h2(const float* __restrict__ agg1, const float* __restrict__ b1,
                         const int* __restrict__ x, const float* __restrict__ ent_emb,
                         float* __restrict__ h2) {
  const int n = blockIdx.x, c = threadIdx.x;       // blockDim = 576
  float v = (c < 512) ? elu1(agg1[(size_t)n * 512 + c] + b1[c])
                      : ent_emb[(size_t)x[n] * 64 + (c - 512)];
  h2[(size_t)n * 576 + c] = v;
}

__global__ void logits2(const float* __restrict__ hl2, const float* __restrict__ as_,
                        const float* __restrict__ ad_, float* __restrict__ als,
                        float* __restrict__ ald, int N) {
  int n = blockIdx.x * blockDim.x + threadIdx.x;
  if (n >= N) return;
  const float* row = hl2 + (size_t)n * 64;
  float ss = 0.f, sd = 0.f;
#pragma unroll 8
  for (int c = 0; c < 64; ++c) { float v = row[c]; ss += v * as_[c]; sd += v * ad_[c]; }
  als[n] = ss; ald[n] = sd;
}

__global__ void edge2_logits(const int* __restrict__ ei, int E, int E2,
                             const float* __restrict__ als, const float* __restrict__ ald,
                             float* __restrict__ eb, unsigned* __restrict__ m2) {
  int e = blockIdx.x * blockDim.x + threadIdx.x;
  if (e >= E2) return;
  int s, d; edge_sd(ei, E, e, s, d);
  float v = lrelu(als[s] + ald[d]);
  eb[e] = v;
  atomicMax(&m2[d], ford(v));
}

__global__ void edge2_expsum(const int* __restrict__ ei, int E, int E2,
                             float* __restrict__ eb, const unsigned* __restrict__ m2,
                             float* __restrict__ s2) {
  int e = blockIdx.x * blockDim.x + threadIdx.x;
  if (e >= E2) return;
  int s, d; edge_sd(ei, E, e, s, d);
  float ex = __expf(eb[e] - unford(m2[d]));
  eb[e] = ex;
  atomicAdd(&s2[d], ex);
}

__global__ void edge2_agg(const int* __restrict__ ei, int E, int E2,
                          const float* __restrict__ eb, const float* __restrict__ s2,
                          const float* __restrict__ hl2, float* __restrict__ agg2) {
  int idx = blockIdx.x * blockDim.x + threadIdx.x;
  if (idx >= E2 * 64) return;
  const int e = idx >> 6, c = idx & 63;
  int s, d; edge_sd(ei, E, e, s, d);
  const float alpha = eb[e] / (s2[d] + 1e-16f);
  atomicAdd(&agg2[(size_t)d * 64 + c], hl2[(size_t)s * 64 + c] * alpha);
}

// g[c] += (1/N) * sum_n elu(agg2[n,c] + b2[c])  (block-local LDS reduce first)
__global__ void __launch_bounds__(256) mean_reduce(
    const float* __restrict__ agg2, const float* __restrict__ b2,
    float* __restrict__ g, int N) {
  __shared__ float red[256];
  const int c = threadIdx.x & 63;
  const int grp = threadIdx.x >> 6;                // 0..3
  const float bc = b2[c];
  float acc = 0.f;
  for (int n = blockIdx.x * 4 + grp; n < N; n += gridDim.x * 4)
    acc += elu1(agg2[(size_t)n * 64 + c] + bc);
  red[threadIdx.x] = acc;
  __syncthreads();
  if (threadIdx.x < 64) {
    float s = red[c] + red[c + 64] + red[c + 128] + red[c + 192];
    atomicAdd(&g[c], s * (1.0f / (float)N));
  }
}

__global__ void final_fc(const float* __restrict__ g, const float* __restrict__ fc_w,
                         const float* __restrict__ fc_b, float* __restrict__ out, int J) {
  __shared__ float gs[64];
  if (threadIdx.x < 64) gs[threadIdx.x] = g[threadIdx.x];
  __syncthreads();
  int j = blockIdx.x * blockDim.x + threadIdx.x;
  if (j >= J) return;
  float s = fc_b[j];
#pragma unroll 8
  for (int c = 0; c < 64; ++c) s += gs[c] * fc_w[c * J + j];
  out[j] = s;
}

// ---------------------------------------------------------------------------
extern "C" void kernel_launch(void* const* d_in, const int* in_sizes, int n_in,
                              void* d_out, int out_size, void* d_ws, size_t ws_size,
                              hipStream_t stream) {
  (void)n_in; (void)ws_size;
  const int*   x       = (const int*)d_in[0];
  const int*   ei      = (const int*)d_in[1];
  const int*   et      = (const int*)d_in[2];
  const float* ent_emb = (const float*)d_in[3];
  const float* rel_emb = (const float*)d_in[4];
  const float* W1      = (const float*)d_in[5];
  const float* a1s     = (const float*)d_in[6];
  const float* a1d     = (const float*)d_in[7];
  const float* b1      = (const float*)d_in[8];
  const float* W2      = (const float*)d_in[9];
  const float* a2s     = (const float*)d_in[10];
  const float* a2d     = (const float*)d_in[11];
  const float* b2      = (const float*)d_in[12];
  const float* fc_w    = (const float*)d_in[13];
  const float* fc_b    = (const float*)d_in[14];
  float* out = (float*)d_out;

  const int N  = in_sizes[0];        // 30000
  const int E  = in_sizes[1] / 2;    // 240000
  const int E2 = E + N;              // + self loops
  const int J  = out_size;           // 200

  // ---- workspace carve (floats). Region0 [0, N*640) holds h+hl, then is
  // reused for h2 (N*576) once layer-1 aggregation is done. Total ~168 MB.
  float* ws = (float*)d_ws;
  float*    h    = ws;                         // N*128
  float*    hl   = ws + (size_t)N * 128;       // N*512
  float*    h2   = ws;                         // N*576 (reuses h+hl region)
  size_t    off  = (size_t)N * 640;
  float*    als  = ws + off; off += (size_t)N * 8;
  float*    ald  = ws + off; off += (size_t)N * 8;
  float*    e1   = ws + off; off += (size_t)E2 * 8;
  unsigned* m1   = (unsigned*)(ws + off); off += (size_t)N * 8;
  float*    s1   = ws + off; off += (size_t)N * 8;
  float*    agg1 = ws + off; off += (size_t)N * 512;
  float*    hl2  = ws + off; off += (size_t)N * 64;
  float*    al2s = ws + off; off += (size_t)N;
  float*    al2d = ws + off; off += (size_t)N;
  float*    e2   = ws + off; off += (size_t)E2;
  unsigned* m2   = (unsigned*)(ws + off); off += (size_t)N;
  float*    s2   = ws + off; off += (size_t)N;
  float*    agg2 = ws + off; off += (size_t)N * 64;
  float*    g    = ws + off; off += 64;

  // ---- init (graph-capture legal)
  const unsigned NEG_INF_ORD = 0x007FFFFFu;    // ford(-inf)
  fill_u32<<<(N * 8 + 255) / 256, 256, 0, stream>>>(m1, NEG_INF_ORD, N * 8);
  fill_u32<<<(N + 255) / 256, 256, 0, stream>>>(m2, NEG_INF_ORD, N);
  hipMemsetAsync(s1,   0, (size_t)N * 8 * sizeof(float),   stream);
  hipMemsetAsync(agg1, 0, (size_t)N * 512 * sizeof(float), stream);
  hipMemsetAsync(s2,   0, (size_t)N * sizeof(float),       stream);
  hipMemsetAsync(agg2, 0, (size_t)N * 64 * sizeof(float),  stream);
  hipMemsetAsync(g,    0, 64 * sizeof(float),              stream);

  // ---- stage 1: gather + GEMM1 (WMMA f32) + logits
  build_h<<<N, 128, 0, stream>>>(x, et, ent_emb, rel_emb, h);
  {
    const int tiles_m = (N + 15) / 16, tiles_n = 512 / 16;
    const int tiles = tiles_m * tiles_n;
    wmma_gemm_f32<<<(tiles + 7) / 8, 256, 0, stream>>>(h, W1, hl, N, 512, 128,
                                                       tiles_n, tiles);
  }
  logits1<<<(N * 8 + 255) / 256, 256, 0, stream>>>(hl, a1s, a1d, als, ald, N);

  // ---- layer-1 edge softmax + aggregation
  edge1_logits<<<(E2 + 255) / 256, 256, 0, stream>>>(ei, E, E2, als, ald, e1, m1);
  edge1_expsum<<<(E2 + 255) / 256, 256, 0, stream>>>(ei, E, E2, e1, m1, s1);
  edge1_agg<<<E2, 256, 0, stream>>>(ei, E, e1, s1, hl, agg1);

  // ---- stage 2: concat + GEMM2 (WMMA f32) + logits
  build_h2<<<N, 576, 0, stream>>>(agg1, b1, x, ent_emb, h2);
  {
    const int tiles_m = (N + 15) / 16, tiles_n = 64 / 16;
    const int tiles = tiles_m * tiles_n;
    wmma_gemm_f32<<<(tiles + 7) / 8, 256, 0, stream>>>(h2, W2, hl2, N, 64, 576,
                                                       tiles_n, tiles);
  }
  logits2<<<(N + 255) / 256, 256, 0, stream>>>(hl2, a2s, a2d, al2s, al2d, N);

  // ---- layer-2 edge softmax + aggregation
  edge2_logits<<<(E2 + 255) / 256, 256, 0, stream>>>(ei, E, E2, al2s, al2d, e2, m2);
  edge2_expsum<<<(E2 + 255) / 256, 256, 0, stream>>>(ei, E, E2, e2, m2, s2);
  edge2_agg<<<(E2 * 64 + 255) / 256, 256, 0, stream>>>(ei, E, E2, e2, s2, hl2, agg2);

  // ---- mean over nodes + final FC
  mean_reduce<<<1024, 256, 0, stream>>>(agg2, b2, g, N);
  final_fc<<<1, 256, 0, stream>>>(g, fc_w, fc_b, out, J);
}